// MPSVariationalQuantumEigensolver_18356690223120
// MI455X (gfx1250) — compile-verified
//
#include <hip/hip_runtime.h>
#include <hip/hip_bf16.h>

typedef __attribute__((ext_vector_type(2))) float v2f;
typedef __attribute__((ext_vector_type(4))) float v4f;
typedef __attribute__((ext_vector_type(8))) float v8f;

#define DIM 4096
#define BOND 8
#define NMID 10

// ---------------------------------------------------------------------------
// Kernel 1: build psi.  psi[j] = core0[b0,:] · M1[:,p1,:] · ... · core_last[:,p_last]
// Bit mapping: b0 = j>>11, mid core i uses bit (10-i), last uses bit 0.
// Work per thread ~1.3 KFLOP; cores (5KB) are broadcast / L2-hot.
// ---------------------------------------------------------------------------
__global__ void build_psi_kernel(const float* __restrict__ core0,
                                 const float* __restrict__ cores_mid,
                                 const float* __restrict__ core_last,
                                 float* __restrict__ psi) {
    int j = blockIdx.x * blockDim.x + threadIdx.x;
    if (j >= DIM) return;

    float v[BOND];
    int p0 = (j >> 11) & 1;
#pragma unroll
    for (int a = 0; a < BOND; ++a) v[a] = core0[p0 * BOND + a];

#pragma unroll
    for (int i = 0; i < NMID; ++i) {
        int p = (j >> (10 - i)) & 1;
        float w[BOND];
#pragma unroll
        for (int b = 0; b < BOND; ++b) w[b] = 0.0f;
#pragma unroll
        for (int a = 0; a < BOND; ++a) {
            const float* row = cores_mid + ((i * 16 + a * 2 + p) * BOND);
            float va = v[a];
#pragma unroll
            for (int b = 0; b < BOND; ++b) w[b] = fmaf(va, row[b], w[b]);
        }
#pragma unroll
        for (int b = 0; b < BOND; ++b) v[b] = w[b];
    }

    int pl = j & 1;
    float s = 0.0f;
#pragma unroll
    for (int a = 0; a < BOND; ++a) s = fmaf(v[a], core_last[a * 2 + pl], s);
    psi[j] = s;
}

// ---------------------------------------------------------------------------
// Kernel 2: hpsi = H @ psi via V_WMMA_F32_16X16X4_F32.
// Block = 16 rows of H. 8 waves per block split K=4096 into 512-wide chunks.
// A (16x4 f32) layout: lanes 0-15 -> M=lane, VGPR0/1 = K0/K1;
//                      lanes 16-31 -> M=lane-16, VGPR0/1 = K2/K3.
// We map WMMA K {0,1,2,3} -> H columns {c, c+1, c+4, c+5} so each lane does a
// single b128 load (lanes<16 read cols c..c+3, lanes>=16 read c+4..c+7); B is
// loaded from psi with the mirrored layout (broadcast across N).
// Each unrolled iteration consumes two 128B cachelines (256B) per row.
// ---------------------------------------------------------------------------
__global__ void gemv_wmma_kernel(const float* __restrict__ H,
                                 const float* __restrict__ psi,
                                 float* __restrict__ hpsi) {
    const int lane = threadIdx.x & 31;
    const int wave = threadIdx.x >> 5;          // 0..7
    const int half4 = (lane >> 4) * 4;          // 0 for lanes 0-15, 4 for 16-31
    const int m = lane & 15;
    const int rowbase = blockIdx.x * 16;
    const float* Hrow = H + (size_t)(rowbase + m) * DIM;

    const int kbase = wave * 512;
    const int kend = kbase + 512;

    v8f acc0 = {};
    v8f acc1 = {};

    for (int k0 = kbase; k0 < kend; k0 += 64) {
        v4f ha[8], pb[8];
#pragma unroll
        for (int u = 0; u < 8; ++u) {
            int c = k0 + 8 * u + half4;
            ha[u] = __builtin_nontemporal_load((const v4f*)(Hrow + c)); // H read once: NT
            pb[u] = *(const v4f*)(psi + c);                             // psi: cache-hot
        }
#pragma unroll
        for (int u = 0; u < 8; ++u) {
            v2f a0 = {ha[u].x, ha[u].y};
            v2f b0 = {pb[u].x, pb[u].y};
            v2f a1 = {ha[u].z, ha[u].w};
            v2f b1 = {pb[u].z, pb[u].w};
            acc0 = __builtin_amdgcn_wmma_f32_16x16x4_f32(
                false, a0, false, b0, (short)0, acc0, false, false);
            acc1 = __builtin_amdgcn_wmma_f32_16x16x4_f32(
                false, a1, false, b1, (short)0, acc1, false, false);
        }
    }

    // Column 0 of D holds the GEMV result for this wave's K-chunk.
    // C/D layout: VGPR r, lane 0 -> (M=r, N=0); lane 16 -> (M=8+r, N=0).
    float accs[8];
#pragma unroll
    for (int r = 0; r < 8; ++r) accs[r] = acc0[r] + acc1[r];

    __shared__ float red[8][16];
    if (lane == 0) {
#pragma unroll
        for (int r = 0; r < 8; ++r) red[wave][r] = accs[r];
    }
    if (lane == 16) {
#pragma unroll
        for (int r = 0; r < 8; ++r) red[wave][8 + r] = accs[r];
    }
    __syncthreads();

    if (threadIdx.x < 16) {
        float s = 0.0f;
#pragma unroll
        for (int w = 0; w < 8; ++w) s += red[w][threadIdx.x];
        hpsi[rowbase + threadIdx.x] = s;
    }
}

// ---------------------------------------------------------------------------
// Kernel 3: E = (psi.hpsi)/(psi.psi); Var = max(|hpsi|^2/(psi.psi) - E^2, 0).
// Uses psi^T H^2 psi == ||H psi||^2 (H symmetric) -> only one pass over H.
// ---------------------------------------------------------------------------
__global__ void finalize_kernel(const float* __restrict__ psi,
                                const float* __restrict__ hpsi,
                                float* __restrict__ out) {
    __shared__ float spp[256], sph[256], shh[256];
    int t = threadIdx.x;
    float pp = 0.0f, ph = 0.0f, hh = 0.0f;
    for (int i = t; i < DIM; i += 256) {
        float p = psi[i];
        float h = hpsi[i];
        pp = fmaf(p, p, pp);
        ph = fmaf(p, h, ph);
        hh = fmaf(h, h, hh);
    }
    spp[t] = pp; sph[t] = ph; shh[t] = hh;
    __syncthreads();
    for (int s = 128; s > 0; s >>= 1) {
        if (t < s) {
            spp[t] += spp[t + s];
            sph[t] += sph[t + s];
            shh[t] += shh[t + s];
        }
        __syncthreads();
    }
    if (t == 0) {
        float norm = spp[0];
        float e = sph[0] / norm;
        float e2 = shh[0] / norm;
        float var = e2 - e * e;
        if (var < 0.0f) var = 0.0f;
        out[0] = e;
        out[1] = var;
    }
}

extern "C" void kernel_launch(void* const* d_in, const int* in_sizes, int n_in,
                              void* d_out, int out_size, void* d_ws, size_t ws_size,
                              hipStream_t stream) {
    const float* core0     = (const float*)d_in[0]; // (2, 8)
    const float* cores_mid = (const float*)d_in[1]; // (10, 8, 2, 8)
    const float* core_last = (const float*)d_in[2]; // (8, 2)
    const float* H         = (const float*)d_in[3]; // (4096, 4096)
    float* out = (float*)d_out;                     // [energy, variance]

    float* psi  = (float*)d_ws;  // 4096 floats
    float* hpsi = psi + DIM;     // 4096 floats

    build_psi_kernel<<<DIM / 256, 256, 0, stream>>>(core0, cores_mid, core_last, psi);
    gemv_wmma_kernel<<<DIM / 16, 256, 0, stream>>>(H, psi, hpsi);
    finalize_kernel<<<1, 256, 0, stream>>>(psi, hpsi, out);
}